// NeighborhoodFusionLayer_90391881711710
// MI455X (gfx1250) — compile-verified
//
#include <hip/hip_runtime.h>
#include <math.h>

typedef float v2f __attribute__((ext_vector_type(2)));
typedef float v8f __attribute__((ext_vector_type(8)));

#define HIDDEN 128
#define HEADS  8

// ---------------------------------------------------------------- utilities
__global__ void zero_f32_kernel(float* __restrict__ p, long long n) {
  long long i = (long long)blockIdx.x * blockDim.x + threadIdx.x;
  long long stride = (long long)gridDim.x * blockDim.x;
  for (; i < n; i += stride) p[i] = 0.0f;
}

// Pack head-mean of W straight into the per-lane WMMA B-fragment layout:
//   p = ((j*32 + k)*32 + lane)*2 + c  ->  Wmean[4k + 2*(lane>>4) + c][16j + (lane&15)]
// so in the GEMM each lane does ONE coalesced v2f load per wmma.
// Also: b_mean[128] and softmax(hop_weights) -> wsm[2].
__global__ void prep_kernel(const float* __restrict__ W, const float* __restrict__ b,
                            const float* __restrict__ hw,
                            float* __restrict__ Bpack, float* __restrict__ bm,
                            float* __restrict__ wsm) {
  int t = blockIdx.x * blockDim.x + threadIdx.x;
  int stride = gridDim.x * blockDim.x;
  for (int p = t; p < HIDDEN * HIDDEN; p += stride) {
    int c    = p & 1;
    int lane = (p >> 1) & 31;
    int k    = (p >> 6) & 31;
    int j    = p >> 11;
    int kk   = 4 * k + ((lane >> 4) << 1) + c;
    int n    = 16 * j + (lane & 15);
    float s = 0.f;
#pragma unroll
    for (int h = 0; h < HEADS; ++h) s += W[h * HIDDEN * HIDDEN + kk * HIDDEN + n];
    Bpack[p] = s * (1.0f / HEADS);
  }
  for (int i = t; i < HIDDEN; i += stride) {
    float s = 0.f;
#pragma unroll
    for (int h = 0; h < HEADS; ++h) s += b[h * HIDDEN + i];
    bm[i] = s * (1.0f / HEADS);
  }
  if (t == 0) {
    float a0 = hw[0], a1 = hw[1];
    float m  = fmaxf(a0, a1);
    float e0 = expf(a0 - m), e1 = expf(a1 - m);
    float inv = 1.0f / (e0 + e1);
    wsm[0] = e0 * inv;
    wsm[1] = e1 * inv;
  }
}

__global__ void deg_kernel(const int* __restrict__ dst, float* __restrict__ deg, int E) {
  int i = blockIdx.x * blockDim.x + threadIdx.x;
  int stride = gridDim.x * blockDim.x;
  for (; i < E; i += stride) atomicAdd(&deg[dst[i]], 1.0f);
}

__global__ void invdeg_kernel(const float* __restrict__ deg, float* __restrict__ inv, int N) {
  int i = blockIdx.x * blockDim.x + threadIdx.x;
  if (i < N) {
    float d = deg[i];
    inv[i] = (d > 0.f) ? (1.0f / d) : 0.0f;
  }
}

// ---------------------------------------------------------------- WMMA GEMM
// Out[N,128] = H[N,128] @ Wmean + bmean  (f32, V_WMMA_F32_16X16X4_F32).
// Block = 128 threads (4 waves). The 64KB pre-packed B is async-copied into
// LDS once per workgroup (global_load_async_to_lds_b128 / ASYNCcnt), then
// each wave computes a 16-row strip, reading B fragments via ds_load_b64.
__global__ void __launch_bounds__(128)
gemm_wmma_kernel(const float* __restrict__ H, const float* __restrict__ Bpack,
                 const float* __restrict__ bm, float* __restrict__ Out, int N) {
  __shared__ float Bs[HIDDEN * HIDDEN];      // 64 KB packed B fragments

  // ---- cooperative async fill: 128 threads x 32 chunks x 16B = 64 KB
  {
    const unsigned ldsBase = (unsigned)(size_t)(void*)&Bs[0];
    const int t = threadIdx.x;
#pragma unroll
    for (int i = 0; i < 32; ++i) {
      unsigned byteOff = (unsigned)((t + i * 128) * 16);
      unsigned ldsAddr = ldsBase + byteOff;
      asm volatile("global_load_async_to_lds_b128 %0, %1, %2"
                   :: "v"(ldsAddr), "v"(byteOff), "s"(Bpack)
                   : "memory");
    }
    asm volatile("s_wait_asynccnt 0x0" ::: "memory");
  }
  __syncthreads();

  const int lane    = threadIdx.x & 31;
  const int wave    = (blockIdx.x * blockDim.x + threadIdx.x) >> 5;
  const int r0      = wave * 16;
  if (r0 >= N) return;                       // wave-uniform exit (EXEC stays full)
  const int mRow    = lane & 15;
  const int hiHalf  = lane >> 4;             // 0 or 1
  const int kOff    = hiHalf * 2;            // 0 or 2

  int row = r0 + mRow;
  if (row >= N) row = N - 1;                 // clamp loads on tail strip
  const float* hrow = H + (size_t)row * HIDDEN + kOff;

  v2f a[32];                                 // full K=128 strip of A fragments
#pragma unroll
  for (int k = 0; k < 32; ++k) a[k] = *(const v2f*)(hrow + 4 * k);

  const v2f* Bsv = (const v2f*)Bs;

#pragma unroll 1
  for (int j = 0; j < 8; ++j) {              // 8 column tiles of 16
    const int n = 16 * j + mRow;
    const float bias = bm[n];
    v8f acc = {bias, bias, bias, bias, bias, bias, bias, bias};
    const v2f* bptr = Bsv + j * (32 * 32) + lane;
#pragma unroll
    for (int k = 0; k < 32; ++k) {
      v2f bf = bptr[k * 32];                 // ds_load_b64, coalesced
      acc = __builtin_amdgcn_wmma_f32_16x16x4_f32(
          /*neg_a=*/false, a[k], /*neg_b=*/false, bf,
          /*c_mod=*/(short)0, acc, /*reuse_a=*/false, /*reuse_b=*/false);
    }
#pragma unroll
    for (int r = 0; r < 8; ++r) {
      const int outRow = r0 + r + hiHalf * 8;
      if (outRow < N) Out[(size_t)outRow * HIDDEN + n] = acc[r];
    }
  }
}

// ---------------------------------------------------------------- edge phase
// 32 threads per edge, 4 columns each: agg[dst] += ht[src]
__global__ void scatter_kernel(const float* __restrict__ ht, const int* __restrict__ src,
                               const int* __restrict__ dst, float* __restrict__ agg, int E) {
  long long i = (long long)blockIdx.x * blockDim.x + threadIdx.x;
  long long stride = (long long)gridDim.x * blockDim.x;
  const long long total = (long long)E * 32;
  for (; i < total; i += stride) {
    int e  = (int)(i >> 5);
    int c4 = (int)(i & 31) * 4;
    int s = src[e], d = dst[e];
    const float4 v = *(const float4*)(ht + (size_t)s * HIDDEN + c4);
    float* dp = agg + (size_t)d * HIDDEN + c4;
    atomicAdd(dp + 0, v.x);
    atomicAdd(dp + 1, v.y);
    atomicAdd(dp + 2, v.z);
    atomicAdd(dp + 3, v.w);
  }
}

// h = agg * inv_deg ; optionally stash h for next hop ; out (=/+)= w[widx] * h
__global__ void finalize_kernel(const float* __restrict__ agg, const float* __restrict__ inv,
                                const float* __restrict__ wsm, int widx,
                                float* __restrict__ out, float* __restrict__ hnext,
                                int accumulate, int N) {
  long long i = (long long)blockIdx.x * blockDim.x + threadIdx.x;
  long long stride = (long long)gridDim.x * blockDim.x;
  const long long total = (long long)N * HIDDEN;
  const float w = wsm[widx];
  for (; i < total; i += stride) {
    int row = (int)(i >> 7);
    float v = agg[i] * inv[row];
    if (hnext) hnext[i] = v;
    float c = w * v;
    if (accumulate) out[i] += c; else out[i] = c;
  }
}

// ---------------------------------------------------------------- launch
extern "C" void kernel_launch(void* const* d_in, const int* in_sizes, int n_in,
                              void* d_out, int out_size, void* d_ws, size_t ws_size,
                              hipStream_t stream) {
  const float* node = (const float*)d_in[0];   // [N,128]
  const float* W    = (const float*)d_in[1];   // [8,128,128]
  const float* b    = (const float*)d_in[2];   // [8,128]
  const float* hw   = (const float*)d_in[3];   // [2]
  const int*   src  = (const int*)d_in[4];     // [E]
  const int*   dst  = (const int*)d_in[5];     // [E]
  float* out = (float*)d_out;

  const int N = in_sizes[0] / HIDDEN;          // 50000
  const int E = in_sizes[4];                   // 800000

  float* ws    = (float*)d_ws;
  float* Bpack = ws;                           // 16384 (packed W_mean)
  float* bm    = ws + 16384;                   // 128
  float* wsm   = ws + 16512;                   // 2
  float* deg   = ws + 16640;                   // N
  size_t npad = (size_t)((N + 127) & ~127);
  float* inv = deg + npad;                     // N
  size_t base = 16640 + 2 * npad;
  base = (base + 127) & ~(size_t)127;
  float* ht   = ws + base;                     // N*128
  float* agg  = ht  + (size_t)N * HIDDEN;      // N*128
  float* hbuf = agg + (size_t)N * HIDDEN;      // N*128

  const int T = 256;
  const long long nd = (long long)N * HIDDEN;
  const int gz  = (int)((nd + T - 1) / T);
  const int gn  = (N + T - 1) / T;
  const int ge  = (E + T - 1) / T;
  const long long scTotal = (long long)E * 32;
  const int gsc = (int)((scTotal + T - 1) / T);
  const int waves = (N + 15) / 16;
  const int gemmBlocks = (waves + 3) / 4;      // 4 waves (128 threads) per block

  // setup
  zero_f32_kernel<<<gn, T, 0, stream>>>(deg, N);
  zero_f32_kernel<<<gz, T, 0, stream>>>(agg, nd);
  prep_kernel<<<64, T, 0, stream>>>(W, b, hw, Bpack, bm, wsm);
  deg_kernel<<<ge, T, 0, stream>>>(dst, deg, E);
  invdeg_kernel<<<gn, T, 0, stream>>>(deg, inv, N);

  // hop 0
  gemm_wmma_kernel<<<gemmBlocks, 128, 0, stream>>>(node, Bpack, bm, ht, N);
  scatter_kernel<<<gsc, T, 0, stream>>>(ht, src, dst, agg, E);
  finalize_kernel<<<gz, T, 0, stream>>>(agg, inv, wsm, 0, out, hbuf, /*accum=*/0, N);

  // hop 1
  zero_f32_kernel<<<gz, T, 0, stream>>>(agg, nd);
  gemm_wmma_kernel<<<gemmBlocks, 128, 0, stream>>>(hbuf, Bpack, bm, ht, N);
  scatter_kernel<<<gsc, T, 0, stream>>>(ht, src, dst, agg, E);
  finalize_kernel<<<gz, T, 0, stream>>>(agg, inv, wsm, 1, out, (float*)nullptr, /*accum=*/1, N);
}